// GraphAttentionLayer_18236431139061
// MI455X (gfx1250) — compile-verified
//
#include <hip/hip_runtime.h>
#include <hip/hip_bf16.h>
#include <math.h>

// ---------------------------------------------------------------------------
// GAT layer for MI455X (gfx1250, wave32).
//   h_prime = h @ W.T        -> V_WMMA_F32_16X16X4_F32 (full fp32 matrix pipe)
//   edge softmax + scatter   -> L2-resident atomics (working set ~60MB << 192MB L2)
// ---------------------------------------------------------------------------

#define GAT_N 50000
#define GAT_E 1600000
#define GAT_IN 128
#define GAT_OUT 64
#define GAT_SLOPE 0.2f
#define GAT_EPS 1e-8f

typedef float v2f __attribute__((ext_vector_type(2)));
typedef float v8f __attribute__((ext_vector_type(8)));

// ---------------------------------------------------------------------------
// Kernel 0: zero scratch accumulators (e_max, sum_exp, h_new)
// ---------------------------------------------------------------------------
__global__ void gat_zero(float* __restrict__ p, int n) {
  int i = blockIdx.x * blockDim.x + threadIdx.x;
  if (i < n) p[i] = 0.0f;
}

// ---------------------------------------------------------------------------
// Kernel 1: h_prime = h @ W.T  via V_WMMA_F32_16X16X4_F32.
// Block = 128 threads = 4 waves; each wave owns one 16x16 output tile:
//   block b -> rows [16b, 16b+16); wave w -> cols [16w, 16w+16)  (OUT=64=4*16)
// W (64x128 fp32, 32KB) is staged once into LDS per block.
//
// VGPR layouts (ISA 7.12.2, wave32):
//   A (16x4, 2 VGPRs): lanes 0-15 hold M=lane, K = k+{0,1}; lanes 16-31 K = k+{2,3}
//   B (4x16, 2 VGPRs): lanes 0-15 hold N=lane, K = k+{0,1}; lanes 16-31 K = k+{2,3}
//   C/D (16x16, 8 VGPRs): vgpr i -> M = i (lanes 0-15) / i+8 (lanes 16-31), N = lane&15
// ---------------------------------------------------------------------------
__global__ void gat_gemm_wmma(const float* __restrict__ h,
                              const float* __restrict__ W,
                              float* __restrict__ hp) {
  __shared__ float sW[GAT_OUT * GAT_IN]; // 32 KB, W row-major [64][128]

  // cooperative W load: 8192 floats = 2048 float4 across 128 threads
  {
    const float4* Wv = (const float4*)W;
    float4* sWv = (float4*)sW;
    for (int i = threadIdx.x; i < (GAT_OUT * GAT_IN) / 4; i += blockDim.x)
      sWv[i] = Wv[i];
  }
  __syncthreads();

  const int wave = threadIdx.x >> 5;
  const int lane = threadIdx.x & 31;
  const int l16  = lane & 15;
  const int half = (lane >> 4) << 1;      // 0 for lanes 0-15, 2 for lanes 16-31

  const int m0 = blockIdx.x * 16;         // N = 50000 = 3125*16, exact
  const int n0 = wave * 16;

  const float* __restrict__ hrow = h + (size_t)(m0 + l16) * GAT_IN;
  const float* __restrict__ wrow = sW + (n0 + l16) * GAT_IN;

  v8f c = {0.f, 0.f, 0.f, 0.f, 0.f, 0.f, 0.f, 0.f};

#pragma unroll 4
  for (int k = 0; k < GAT_IN; k += 4) {
    float2 av = *(const float2*)(hrow + k + half);  // A[M][k+half .. +1]
    float2 bv = *(const float2*)(wrow + k + half);  // B[k+half][N] = W[N][k+half]
    v2f a = {av.x, av.y};
    v2f b = {bv.x, bv.y};
    c = __builtin_amdgcn_wmma_f32_16x16x4_f32(
        /*neg_a=*/false, a, /*neg_b=*/false, b,
        /*c_mod=*/(short)0, c, /*reuse_a=*/false, /*reuse_b=*/false);
  }

  // scatter D: vgpr i -> row m0 + i + 8*(lane>=16), col n0 + (lane&15)
  const int mbase = m0 + ((lane >> 4) << 3);
  const int col = n0 + l16;
#pragma unroll
  for (int i = 0; i < 8; ++i)
    hp[(size_t)(mbase + i) * GAT_OUT + col] = c[i];
}

// ---------------------------------------------------------------------------
// Kernel 2: per-node attention scores  s_src = hp . a[:64], s_dst = hp . a[64:]
// ---------------------------------------------------------------------------
__global__ void gat_scores(const float* __restrict__ hp,
                           const float* __restrict__ a,
                           float* __restrict__ s_src,
                           float* __restrict__ s_dst) {
  int n = blockIdx.x * blockDim.x + threadIdx.x;
  if (n >= GAT_N) return;
  const float* __restrict__ row = hp + (size_t)n * GAT_OUT;
  float ss = 0.f, sd = 0.f;
#pragma unroll 8
  for (int d = 0; d < GAT_OUT; ++d) {
    float v = row[d];
    ss = fmaf(v, a[d], ss);
    sd = fmaf(v, a[GAT_OUT + d], sd);
  }
  s_src[n] = ss;
  s_dst[n] = sd;
}

// ---------------------------------------------------------------------------
// Kernel 3: edge scores + segment max.
//   e = leaky_relu(s_src[src]+s_dst[dst]) + log1p(max(w,0))
//   e_max[dst] = max over edges, with 0-init (include_self semantics).
// e_max init is 0.0 and max includes 0 => only positive e can win; positive
// f32 bit patterns are monotone as signed ints => int atomicMax is exact.
// ---------------------------------------------------------------------------
__global__ void gat_edge_score(const int* __restrict__ ei,
                               const float* __restrict__ ew,
                               const float* __restrict__ s_src,
                               const float* __restrict__ s_dst,
                               float* __restrict__ e_buf,
                               float* __restrict__ e_max) {
  int i = blockIdx.x * blockDim.x + threadIdx.x;
  if (i >= GAT_E) return;
  int s = ei[i];
  int d = ei[GAT_E + i];
  float x = s_src[s] + s_dst[d];
  float e = (x > 0.f) ? x : GAT_SLOPE * x;
  float w = ew[i];
  e += log1pf(w > 0.f ? w : 0.f);
  e_buf[i] = e;
  if (e > 0.f)
    atomicMax((int*)(e_max + d), __float_as_int(e));
}

// ---------------------------------------------------------------------------
// Kernel 4: exp_e = exp(e - e_max[dst]); sum_exp[dst] += exp_e
// ---------------------------------------------------------------------------
__global__ void gat_edge_exp(const int* __restrict__ ei,
                             float* __restrict__ e_buf,
                             const float* __restrict__ e_max,
                             float* __restrict__ sum_exp) {
  int i = blockIdx.x * blockDim.x + threadIdx.x;
  if (i >= GAT_E) return;
  int d = ei[GAT_E + i];
  float ex = __expf(e_buf[i] - e_max[d]);
  e_buf[i] = ex;                 // e_buf now holds exp_e
  atomicAdd(sum_exp + d, ex);
}

// ---------------------------------------------------------------------------
// Kernel 5: weighted scatter. One wave32 per edge; lane l handles output
// dims l and l+32. 102M fp32 atomic adds, all L2-resident.
// ---------------------------------------------------------------------------
__global__ void gat_scatter(const int* __restrict__ ei,
                            const float* __restrict__ hp,
                            const float* __restrict__ exp_e,
                            const float* __restrict__ sum_exp,
                            float* __restrict__ h_new) {
  int e = blockIdx.x * (blockDim.x >> 5) + (threadIdx.x >> 5);
  if (e >= GAT_E) return;
  int lane = threadIdx.x & 31;
  int src = ei[e];
  int dst = ei[GAT_E + e];
  float alpha = exp_e[e] / (sum_exp[dst] + GAT_EPS);
  const float* __restrict__ hrow = hp + (size_t)src * GAT_OUT;
  float* __restrict__ orow = h_new + (size_t)dst * GAT_OUT;
  atomicAdd(orow + lane,      alpha * hrow[lane]);
  atomicAdd(orow + lane + 32, alpha * hrow[lane + 32]);
}

// ---------------------------------------------------------------------------
// Kernel 6: out = elu(h_new)
// ---------------------------------------------------------------------------
__global__ void gat_elu(const float* __restrict__ h_new, float* __restrict__ out, int n) {
  int i = blockIdx.x * blockDim.x + threadIdx.x;
  if (i >= n) return;
  float x = h_new[i];
  out[i] = (x > 0.f) ? x : expm1f(x);
}

// ---------------------------------------------------------------------------
extern "C" void kernel_launch(void* const* d_in, const int* in_sizes, int n_in,
                              void* d_out, int out_size, void* d_ws, size_t ws_size,
                              hipStream_t stream) {
  (void)in_sizes; (void)n_in; (void)out_size; (void)ws_size;

  const float* h  = (const float*)d_in[0];   // [N,128]
  const int*   ei = (const int*)d_in[1];     // [2,E] (src row, then dst row)
  const float* ew = (const float*)d_in[2];   // [E]
  const float* W  = (const float*)d_in[3];   // [64,128]
  const float* a  = (const float*)d_in[4];   // [128]
  float* out = (float*)d_out;                // [N,64]

  // workspace layout (floats)
  float* ws = (float*)d_ws;
  size_t off = 0;
  float* hp      = ws + off; off += (size_t)GAT_N * GAT_OUT;  // h_prime
  float* s_src   = ws + off; off += GAT_N;
  float* s_dst   = ws + off; off += GAT_N;
  float* e_buf   = ws + off; off += GAT_E;                    // e, then exp_e
  float* e_max   = ws + off; off += GAT_N;
  float* sum_exp = ws + off; off += GAT_N;
  float* h_new   = ws + off; off += (size_t)GAT_N * GAT_OUT;

  // 0) zero accumulators: e_max, sum_exp, h_new (contiguous: e_max..h_new end)
  {
    int nz = GAT_N + GAT_N + GAT_N * GAT_OUT;
    gat_zero<<<(nz + 255) / 256, 256, 0, stream>>>(e_max, nz);
  }

  // 1) h_prime = h @ W.T   (WMMA f32 16x16x4; N=50000=3125*16 rows)
  gat_gemm_wmma<<<GAT_N / 16, 128, 0, stream>>>(h, W, hp);

  // 2) per-node scores
  gat_scores<<<(GAT_N + 255) / 256, 256, 0, stream>>>(hp, a, s_src, s_dst);

  // 3) edge scores + segment max
  gat_edge_score<<<(GAT_E + 255) / 256, 256, 0, stream>>>(ei, ew, s_src, s_dst,
                                                          e_buf, e_max);

  // 4) exp + segment sum
  gat_edge_exp<<<(GAT_E + 255) / 256, 256, 0, stream>>>(ei, e_buf, e_max, sum_exp);

  // 5) weighted scatter: 1 wave per edge, 8 edges per 256-thread block
  gat_scatter<<<(GAT_E + 7) / 8, 256, 0, stream>>>(ei, hp, e_buf, sum_exp, h_new);

  // 6) ELU
  {
    int n = GAT_N * GAT_OUT;
    gat_elu<<<(n + 255) / 256, 256, 0, stream>>>(h_new, out, n);
  }
}